// UnimodalNormal_82094004896338
// MI455X (gfx1250) — compile-verified
//
#include <hip/hip_runtime.h>
#include <hip/hip_bf16.h>
#include <math.h>

// CDNA5 wave32 vectors
typedef __attribute__((ext_vector_type(2))) float v2f;
typedef __attribute__((ext_vector_type(8))) float v8f;

#define K_CH 2048
#define NCLS 100
#define WAVES_PER_BLOCK 8
#define ROWS_PER_WAVE 16
#define ROWS_PER_BLOCK (WAVES_PER_BLOCK * ROWS_PER_WAVE)

// Fused: [mu_dot, prec_dot] = x_tile @ [w_mu, w_prec] via V_WMMA_F32_16X16X4_F32,
// then tanh / softplus / truncated-normal CDF head, normalized by telescoped row sum.
//
// WMMA roles:
//   A (16x4, f32): row0 = w_mu chunk, row1 = w_prec chunk, rows 2..15 zero.
//       lane m=lane&15 holds row m; half-wave (lane>=16) holds K=2,3 (koff).
//   B (4x16, f32): B[k][n] = x[row_base+n][k0+k]; lane n (and n+16 for K=2,3)
//       loads a float2 of its row -> full cacheline consumption per lane pair.
//   C (16x16, f32): C[0][n] = mu dot, C[1][n] = prec dot -> acc[0], acc[1] at lane n.
__global__ __launch_bounds__(WAVES_PER_BLOCK * 32)
void unimodal_head_kernel(const float* __restrict__ x,
                          const float* __restrict__ w_mu,
                          const float* __restrict__ b_mu,
                          const float* __restrict__ w_prec,
                          const float* __restrict__ b_prec,
                          float* __restrict__ out,
                          int batch) {
    const int lane = threadIdx.x & 31;
    const int wave = threadIdx.x >> 5;
    const int row_base = (blockIdx.x * WAVES_PER_BLOCK + wave) * ROWS_PER_WAVE;
    if (row_base >= batch) return;   // wave-uniform exit

    const int n    = lane & 15;         // N index == row within tile
    const int koff = (lane >> 4) * 2;   // this half-wave covers K = koff, koff+1
    const float msk = (n < 2) ? 1.0f : 0.0f;            // zero A rows 2..15
    const float* wsel = (n == 1) ? w_prec : w_mu;       // branchless, valid addr for all lanes
    const float* wptr = wsel + koff;
    const float* xptr = x + (size_t)(row_base + n) * K_CH + koff;

    v8f acc0 = {0.f,0.f,0.f,0.f,0.f,0.f,0.f,0.f};
    v8f acc1 = {0.f,0.f,0.f,0.f,0.f,0.f,0.f,0.f};

#pragma unroll 4
    for (int k0 = 0; k0 < K_CH; k0 += 8) {
        v2f w0 = *(const v2f*)(wptr + k0);
        v2f b0 = *(const v2f*)(xptr + k0);
        v2f w1 = *(const v2f*)(wptr + k0 + 4);
        v2f b1 = *(const v2f*)(xptr + k0 + 4);
        v2f a0; a0.x = w0.x * msk; a0.y = w0.y * msk;
        v2f a1; a1.x = w1.x * msk; a1.y = w1.y * msk;
        acc0 = __builtin_amdgcn_wmma_f32_16x16x4_f32(
                   false, a0, false, b0, (short)0, acc0, false, false);
        acc1 = __builtin_amdgcn_wmma_f32_16x16x4_f32(
                   false, a1, false, b1, (short)0, acc1, false, false);
    }

    // C[0][n] / C[1][n] live in acc VGPR0/VGPR1 at lanes 0..15
    float dmu = acc0[0] + acc1[0];
    float dpr = acc0[1] + acc1[1];
    // broadcast row-n dots to both half-waves (lane n and lane n+16 share row n)
    dmu = __shfl(dmu, n, 32);
    dpr = __shfl(dpr, n, 32);

    const int row = row_base + n;
    if (row >= batch) return;

    float mu = tanhf(dmu + b_mu[0]);
    float sv = dpr + b_prec[0];
    float sp = (sv > 20.0f) ? sv : log1pf(expf(sv));       // softplus
    float prec = fminf(fmaxf(sp, 0.01f), 100.0f);
    float scale = prec * 0.70710678118654752f;             // (t-mu)/sig * 1/sqrt(2), sig=1/prec

    // row sum telescopes: sum(probs) = cdf(+1) - cdf(-1)
    float chi = 0.5f * (1.0f + erff(( 1.0f - mu) * scale));
    float clo = 0.5f * (1.0f + erff((-1.0f - mu) * scale));
    float inv = 1.0f / (chi - clo);

    const int c0 = (lane >> 4) * (NCLS / 2);               // 0 or 50
    float* outp = out + (size_t)row * NCLS + c0;           // 8B-aligned (400B pitch, 200B half)

    float prev = 0.5f * (1.0f + erff((0.02f * (float)c0 - 1.0f - mu) * scale));
    for (int j = 0; j < NCLS / 2; j += 2) {
        float t1 = 0.02f * (float)(c0 + j + 1) - 1.0f;
        float t2 = 0.02f * (float)(c0 + j + 2) - 1.0f;
        float cd1 = 0.5f * (1.0f + erff((t1 - mu) * scale));
        float cd2 = 0.5f * (1.0f + erff((t2 - mu) * scale));
        v2f o; o.x = (cd1 - prev) * inv; o.y = (cd2 - cd1) * inv;
        *(v2f*)(outp + j) = o;
        prev = cd2;
    }
}

extern "C" void kernel_launch(void* const* d_in, const int* in_sizes, int n_in,
                              void* d_out, int out_size, void* d_ws, size_t ws_size,
                              hipStream_t stream) {
    const float* x      = (const float*)d_in[0];
    const float* w_mu   = (const float*)d_in[1];
    const float* b_mu   = (const float*)d_in[2];
    const float* w_prec = (const float*)d_in[3];
    const float* b_prec = (const float*)d_in[4];
    float* out = (float*)d_out;

    const int in_ch = in_sizes[1];            // 2048 (== K_CH)
    const int batch = in_sizes[0] / in_ch;    // 32768
    const int blocks = (batch + ROWS_PER_BLOCK - 1) / ROWS_PER_BLOCK;

    hipLaunchKernelGGL(unimodal_head_kernel, dim3(blocks), dim3(WAVES_PER_BLOCK * 32),
                       0, stream, x, w_mu, b_mu, w_prec, b_prec, out, batch);
}